// Graph_Decoder_1778116460941
// MI455X (gfx1250) — compile-verified
//
#include <hip/hip_runtime.h>
#include <hip/hip_bf16.h>

#define N_HP  49152
#define N_DST 50000
#define NE    400000
#define T_OUT 4
#define H     64
#define HID   128
#define NROWS (T_OUT * N_DST)      // CSR rows: (tt, dst)
#define NEDGE (T_OUT * NE)         // 1.6M

typedef __attribute__((ext_vector_type(2))) float v2f;
typedef __attribute__((ext_vector_type(8))) float v8f;

// ---------------------------------------------------------------------------
// Degree accumulation: rns[tt][src]++ , rnd[tt][dst]++  (as float)
// ---------------------------------------------------------------------------
__global__ void deg_kernel(const int* __restrict__ src, const int* __restrict__ dst,
                           float* __restrict__ rns, float* __restrict__ rnd) {
    int gid = blockIdx.x * blockDim.x + threadIdx.x;
    if (gid >= NEDGE) return;
    int tt = gid / NE;
    atomicAdd(&rns[tt * N_HP  + src[gid]], 1.0f);
    atomicAdd(&rnd[tt * N_DST + dst[gid]], 1.0f);
}

// In-place deg -> rsqrt(max(deg,1))
__global__ void rsqrt_kernel(float* __restrict__ p, int n) {
    int gid = blockIdx.x * blockDim.x + threadIdx.x;
    if (gid < n) p[gid] = rsqrtf(fmaxf(p[gid], 1.0f));
}

// ---------------------------------------------------------------------------
// Two-level exclusive scan of in-degree counts -> CSR row starts.
// scan1: 256 thr x 4 elem = 1024 rows per block; per-block exclusive scan.
// ---------------------------------------------------------------------------
__global__ __launch_bounds__(256) void scan1_kernel(const float* __restrict__ cnt,
                                                    int* __restrict__ rowStart,
                                                    int* __restrict__ blockSum) {
    __shared__ int sh[256];
    const int tid  = threadIdx.x;
    const int base = blockIdx.x * 1024 + tid * 4;
    int v[4], s = 0;
    #pragma unroll
    for (int i = 0; i < 4; ++i) {
        int idx = base + i;
        v[i] = (idx < NROWS) ? (int)cnt[idx] : 0;
        s += v[i];
    }
    sh[tid] = s;
    __syncthreads();
    for (int off = 1; off < 256; off <<= 1) {        // inclusive Hillis-Steele
        int a = sh[tid];
        int b = (tid >= off) ? sh[tid - off] : 0;
        __syncthreads();
        sh[tid] = a + b;
        __syncthreads();
    }
    if (tid == 255) blockSum[blockIdx.x] = sh[255];
    int run = (tid == 0) ? 0 : sh[tid - 1];
    #pragma unroll
    for (int i = 0; i < 4; ++i) {
        int idx = base + i;
        if (idx < NROWS) rowStart[idx] = run;
        run += v[i];
    }
}

// scan2: exclusive scan of <=256 block sums in place (single block)
__global__ __launch_bounds__(256) void scan2_kernel(int* __restrict__ blockSum, int nb) {
    __shared__ int sh[256];
    const int tid = threadIdx.x;
    sh[tid] = (tid < nb) ? blockSum[tid] : 0;
    __syncthreads();
    for (int off = 1; off < 256; off <<= 1) {
        int a = sh[tid];
        int b = (tid >= off) ? sh[tid - off] : 0;
        __syncthreads();
        sh[tid] = a + b;
        __syncthreads();
    }
    if (tid < nb) blockSum[tid] = (tid == 0) ? 0 : sh[tid - 1];
}

// scan3: add block offsets; duplicate into cursor; append grand total sentinel
__global__ void scan3_kernel(int* __restrict__ rowStart, int* __restrict__ cursor,
                             const int* __restrict__ blockSum) {
    int gid = blockIdx.x * blockDim.x + threadIdx.x;
    if (gid < NROWS) {
        int v = rowStart[gid] + blockSum[gid >> 10];
        rowStart[gid] = v;
        cursor[gid]   = v;
    }
    if (gid == NROWS) rowStart[NROWS] = NEDGE;   // total edge count is static
}

// ---------------------------------------------------------------------------
// CSR fill: edge -> (src, coef = rsqrt(outdeg)*rsqrt(indeg)) bucketed by (tt,dst)
// ---------------------------------------------------------------------------
__global__ void fill_kernel(const int* __restrict__ src, const int* __restrict__ dst,
                            const float* __restrict__ rns, const float* __restrict__ rnd,
                            int* __restrict__ cursor,
                            int* __restrict__ csrSrc, float* __restrict__ csrCoef) {
    int gid = blockIdx.x * blockDim.x + threadIdx.x;
    if (gid >= NEDGE) return;
    int tt = gid / NE;
    int s = src[gid], d = dst[gid];
    int pos = atomicAdd(&cursor[tt * N_DST + d], 1);
    csrSrc[pos]  = s;
    csrCoef[pos] = rns[tt * N_HP + s] * rnd[tt * N_DST + d];
}

// ---------------------------------------------------------------------------
// Y = X @ W  (X: 196608 x 64 node-major all-t features, W: 64x64)
// WMMA f32 16x16x4, K=64 (16 steps). Block = 4 waves; wave w owns N-tile w.
// A strip staged in LDS, padded stride 66 (bank-conflict-free for f32x2 reads).
// ---------------------------------------------------------------------------
__global__ __launch_bounds__(128) void gemm_xw(const float* __restrict__ X,
                                               const float* __restrict__ W,
                                               float* __restrict__ Y) {
    __shared__ float As[16 * 66];
    const int tid  = threadIdx.x;
    const int lane = tid & 31;
    const int wv   = tid >> 5;
    const int m0   = blockIdx.x * 16;   // 12288 blocks * 16 = 196608 exact
    const int n0   = wv * 16;
    const int l15  = lane & 15;
    const int lhi  = lane >> 4;

    {   // stage A tile (16 x 64)
        int row = tid >> 3;
        int c0  = (tid & 7) * 8;
        const float* g = X + ((size_t)(m0 + row)) * H + c0;
        float4 u0 = *(const float4*)(g + 0);
        float4 u1 = *(const float4*)(g + 4);
        float* sm = &As[row * 66 + c0];
        sm[0] = u0.x; sm[1] = u0.y; sm[2] = u0.z; sm[3] = u0.w;
        sm[4] = u1.x; sm[5] = u1.y; sm[6] = u1.z; sm[7] = u1.w;
    }
    __syncthreads();

    v8f dacc = {};
    #pragma unroll
    for (int k0 = 0; k0 < H; k0 += 4) {
        v2f a = *(const v2f*)&As[l15 * 66 + k0 + 2 * lhi];   // A: K pair per lane half
        v2f b;                                               // B: VGPR0=K{0,2}, VGPR1=K{1,3}
        b.x = W[(k0 + 2 * lhi + 0) * H + n0 + l15];
        b.y = W[(k0 + 2 * lhi + 1) * H + n0 + l15];
        dacc = __builtin_amdgcn_wmma_f32_16x16x4_f32(
            false, a, false, b, (short)0, dacc, false, false);
    }
    #pragma unroll
    for (int v = 0; v < 8; ++v) {
        int m = m0 + v + 8 * lhi;        // C/D: VGPR v -> row v (+8 hi lanes)
        Y[(size_t)m * H + n0 + l15] = dacc[v];
    }
}

// ---------------------------------------------------------------------------
// Gather-aggregate (no atomics): one wave32 per destination row of edge-set tt.
// acc[t] = sum_j coef_j * Y[src_j][t][:]   (2 feats/lane, all 4 t per pass)
// space1[t][d][:] += leaky_relu(acc[t] + conv_b[tt])
// ---------------------------------------------------------------------------
__global__ __launch_bounds__(256) void gather_kernel(const float* __restrict__ Y,
        const int* __restrict__ rowStart, const int* __restrict__ csrSrc,
        const float* __restrict__ csrCoef, const float* __restrict__ convB,
        float* __restrict__ space1, int tt) {
    int gid  = blockIdx.x * blockDim.x + threadIdx.x;
    int d    = gid >> 5;                 // 50000 waves, grid exact
    int lane = gid & 31;
    int row  = tt * N_DST + d;
    int j0 = rowStart[row], j1 = rowStart[row + 1];

    float2 acc[T_OUT] = {};
    for (int j = j0; j < j1; ++j) {
        int   s = csrSrc[j];
        float c = csrCoef[j];
        if (j + 1 < j1)
            __builtin_prefetch(Y + (size_t)csrSrc[j + 1] * (T_OUT * H), 0, 1);
        const float* yp = Y + (size_t)s * (T_OUT * H) + lane * 2;
        #pragma unroll
        for (int t = 0; t < T_OUT; ++t) {
            float2 v = *(const float2*)(yp + t * H);
            acc[t].x += c * v.x;
            acc[t].y += c * v.y;
        }
    }
    float b0 = convB[tt * H + lane * 2 + 0];
    float b1 = convB[tt * H + lane * 2 + 1];
    #pragma unroll
    for (int t = 0; t < T_OUT; ++t) {
        float z0 = acc[t].x + b0;
        float z1 = acc[t].y + b1;
        z0 = (z0 > 0.0f) ? z0 : 0.01f * z0;
        z1 = (z1 > 0.0f) ? z1 : 0.01f * z1;
        float* sp = space1 + ((size_t)t * N_DST + d) * H + lane * 2;
        sp[0] += z0;
        sp[1] += z1;
    }
}

// ---------------------------------------------------------------------------
// Fused attention + decode, one block (128 thr) per destination node.
// ---------------------------------------------------------------------------
__global__ __launch_bounds__(128) void attn_kernel(const float* __restrict__ space1,
        const float* __restrict__ prev,
        const float* __restrict__ linW, const float* __restrict__ linB,
        const float* __restrict__ W1,   const float* __restrict__ b1,
        const float* __restrict__ W2,   const float* __restrict__ b2,
        const float* __restrict__ decW, const float* __restrict__ decB,
        float* __restrict__ out) {
    __shared__ float  s1[H], s2[H];
    __shared__ float2 red[128];
    const int d   = blockIdx.x;
    const int tid = threadIdx.x;

    float pv = prev[d];
    if (tid < H) {
        s1[tid] = space1[(size_t)d * H + tid];
        s2[tid] = pv * linW[tid] + linB[tid];
    }
    __syncthreads();

    float h1 = b1[tid], h2 = b1[tid];
    #pragma unroll 8
    for (int k = 0; k < H; ++k) {
        float w = W1[k * HID + tid];
        h1 += s1[k] * w;
        h2 += s2[k] * w;
    }
    h1 = tanhf(h1); h2 = tanhf(h2);
    float w2 = W2[tid];
    red[tid] = make_float2(h1 * w2, h2 * w2);
    __syncthreads();
    for (int off = 64; off > 0; off >>= 1) {
        if (tid < off) {
            red[tid].x += red[tid + off].x;
            red[tid].y += red[tid + off].y;
        }
        __syncthreads();
    }
    float l1 = red[0].x + b2[0];
    float l2 = red[0].y + b2[0];
    float mx = fmaxf(l1, l2);
    float e1 = expf(l1 - mx), e2 = expf(l2 - mx);
    float inv = 1.0f / (e1 + e2);
    float a1 = e1 * inv, a2 = e2 * inv;
    __syncthreads();

    float q = 0.0f;
    if (tid < H) q = (a1 * s1[tid] + a2 * s2[tid]) * decW[tid];
    red[tid].x = q;
    __syncthreads();
    for (int off = 64; off > 0; off >>= 1) {
        if (tid < off) red[tid].x += red[tid + off].x;
        __syncthreads();
    }
    if (tid == 0) out[d] = red[0].x + decB[0];
}

// ---------------------------------------------------------------------------
static inline size_t alignup(size_t v) { return (v + 63) & ~(size_t)63; }  // 64 elems

extern "C" void kernel_launch(void* const* d_in, const int* in_sizes, int n_in,
                              void* d_out, int out_size, void* d_ws, size_t ws_size,
                              hipStream_t stream) {
    const float* x     = (const float*)d_in[0];
    const float* prev0 = (const float*)d_in[1];
    const float* convW = (const float*)d_in[2];
    const float* convB = (const float*)d_in[3];
    const float* linW  = (const float*)d_in[4];
    const float* linB  = (const float*)d_in[5];
    const float* W1    = (const float*)d_in[6];
    const float* b1    = (const float*)d_in[7];
    const float* W2    = (const float*)d_in[8];
    const float* b2    = (const float*)d_in[9];
    const float* decW  = (const float*)d_in[10];
    const float* decB  = (const float*)d_in[11];
    const int* srcIdx  = (const int*)d_in[12];
    const int* dstIdx  = (const int*)d_in[13];
    float* out = (float*)d_out;

    // workspace carve-up (64-element = 256B aligned regions)
    float* base = (float*)d_ws;
    size_t off = 0;
    float* rns      = base + off; off += alignup((size_t)T_OUT * N_HP);        // counts->rsqrt
    float* rnd      = base + off; off += alignup((size_t)T_OUT * N_DST);
    int*   rowStart = (int*)(base + off); off += alignup((size_t)NROWS + 1);
    int*   cursor   = (int*)(base + off); off += alignup((size_t)NROWS);
    int*   blockSum = (int*)(base + off); off += alignup((size_t)256);
    int*   csrSrc   = (int*)(base + off); off += alignup((size_t)NEDGE);
    float* csrCoef  = base + off; off += alignup((size_t)NEDGE);
    float* Y        = base + off; off += alignup((size_t)N_HP * T_OUT * H);    // 50 MB
    float* space1   = base + off; off += alignup((size_t)T_OUT * N_DST * H);   // 51.2 MB

    const int nb1 = (NROWS + 1023) / 1024;   // 196 scan blocks

    // --- degree histograms ---
    hipMemsetAsync(rns, 0, (size_t)(T_OUT * N_HP + 64 + T_OUT * N_DST) * sizeof(float), stream);
    deg_kernel<<<(NEDGE + 255) / 256, 256, 0, stream>>>(srcIdx, dstIdx, rns, rnd);
    // --- CSR row offsets from in-degree counts (reads rnd BEFORE rsqrt) ---
    scan1_kernel<<<nb1, 256, 0, stream>>>(rnd, rowStart, blockSum);
    scan2_kernel<<<1, 256, 0, stream>>>(blockSum, nb1);
    rsqrt_kernel<<<(T_OUT * N_HP  + 255) / 256, 256, 0, stream>>>(rns, T_OUT * N_HP);
    rsqrt_kernel<<<(T_OUT * N_DST + 255) / 256, 256, 0, stream>>>(rnd, T_OUT * N_DST);
    scan3_kernel<<<(NROWS + 1 + 255) / 256, 256, 0, stream>>>(rowStart, cursor, blockSum);
    fill_kernel<<<(NEDGE + 255) / 256, 256, 0, stream>>>(srcIdx, dstIdx, rns, rnd,
                                                         cursor, csrSrc, csrCoef);

    // --- space1[t] for all t: per tt, transform-then-gather (no float atomics) ---
    hipMemsetAsync(space1, 0, (size_t)T_OUT * N_DST * H * sizeof(float), stream);
    for (int tt = 0; tt < T_OUT; ++tt) {
        gemm_xw<<<(N_HP * T_OUT) / 16, 128, 0, stream>>>(x, convW + tt * H * H, Y);
        gather_kernel<<<(N_DST * 32) / 256, 256, 0, stream>>>(
            Y, rowStart, csrSrc, csrCoef, convB, space1, tt);
    }

    // --- sequential attention/decode chain (out[t] is prev for t+1) ---
    for (int t = 0; t < T_OUT; ++t) {
        const float* prev = (t == 0) ? prev0 : (out + (size_t)(t - 1) * N_DST);
        attn_kernel<<<N_DST, 128, 0, stream>>>(space1 + (size_t)t * N_DST * H, prev,
                                               linW, linB, W1, b1, W2, b2, decW, decB,
                                               out + (size_t)t * N_DST);
    }
}